// GraphNodeFeatureEncoder_36756330119411
// MI455X (gfx1250) — compile-verified
//
#include <hip/hip_runtime.h>

typedef __attribute__((ext_vector_type(2))) float v2f;
typedef __attribute__((ext_vector_type(8))) float v8f;

#define ATOMS_PER_MOL 32
#define F_COLS 10
#define VOCAB 4096
#define PER_COL_DIM 32
#define HIDDEN 256
#define SENTINEL 999999999
#define MOLS_PER_WG 16
#define WG_THREADS 256

// Fused: (1) gather+sum 10 embedding tables per atom, (2) mean-pool over the
// molecule (pooling commutes with the linear projection), (3) project the
// pooled 32-vector to 256 dims with V_WMMA_F32_16X16X4_F32 (K=32 in 8 steps).
__global__ __launch_bounds__(WG_THREADS) void GraphNodeFeatureEncoder_kernel(
    const int* __restrict__ a_features,   // [n_atoms][10] int32
    const int* __restrict__ a_scopes,     // [n_mols][2]   int32 (start, len)
    const float* __restrict__ emb,        // [10][4096][32] f32
    const float* __restrict__ W,          // [256][32] f32  (proj_w)
    const float* __restrict__ bias,       // [256] f32
    float* __restrict__ out,              // [n_mols][256] f32
    int n_mols)
{
    __shared__ int   idxL[MOLS_PER_WG][ATOMS_PER_MOL][F_COLS]; // 20 KB
    __shared__ float pooled[MOLS_PER_WG][PER_COL_DIM + 1];     // pad: bank spread
    __shared__ int   startL[MOLS_PER_WG];
    __shared__ int   lenL[MOLS_PER_WG];

    const int tid = threadIdx.x;
    const int m0  = blockIdx.x * MOLS_PER_WG;

    if (tid < MOLS_PER_WG) {
        int m = m0 + tid;
        if (m < n_mols) {
            startL[tid] = a_scopes[m * 2 + 0];
            lenL[tid]   = a_scopes[m * 2 + 1];
        } else {
            startL[tid] = 0;
            lenL[tid]   = 0;
        }
    }
    __syncthreads();

    // ---- Phase 0: stage sentinel-mapped indices into LDS ----
    for (int j = tid; j < MOLS_PER_WG * ATOMS_PER_MOL * F_COLS; j += WG_THREADS) {
        int mol  = j / (ATOMS_PER_MOL * F_COLS);
        int r    = j - mol * (ATOMS_PER_MOL * F_COLS);
        int atom = r / F_COLS;
        int col  = r - atom * F_COLS;
        int id   = 0;
        if (atom < lenL[mol]) {
            int f = a_features[(size_t)(startL[mol] + atom) * F_COLS + col];
            // reference: idx = (f >= SENTINEL) ? 0 : f mod VOCAB (non-negative)
            id = (f >= SENTINEL) ? 0 : (f & (VOCAB - 1));
        }
        idxL[mol][atom][col] = id;
    }
    __syncthreads();

    // ---- Phase 1: pooled mean of summed embeddings ----
    // thread t owns (mol = t/16, dims d0..d0+1); gathers are L2-resident.
    {
        int mol = tid >> 4;
        int d0  = (tid & 15) * 2;
        int len = lenL[mol];
        int na  = len < ATOMS_PER_MOL ? len : ATOMS_PER_MOL;
        float ax = 0.f, ay = 0.f;
        for (int a = 0; a < na; ++a) {
            #pragma unroll
            for (int c = 0; c < F_COLS; ++c) {
                const float* row =
                    emb + ((size_t)(c * VOCAB + idxL[mol][a][c])) * PER_COL_DIM + d0;
                v2f v = *(const v2f*)row;
                ax += v.x;
                ay += v.y;
            }
        }
        float s = (len > 0) ? (1.0f / (float)len) : 0.0f;
        pooled[mol][d0]     = ax * s;
        pooled[mol][d0 + 1] = ay * s;
    }
    __syncthreads();

    // ---- Phase 2: D = pooled(16x32) @ W^T(32x256) + bias via WMMA fp32 ----
    {
        const int wave = tid >> 5;
        const int lane = tid & 31;
        const int mrow = lane & 15;   // M row (A) / N col (B,C,D)
        const int hi   = lane >> 4;   // lanes 16-31 hold K+2 (A,B) / M+8 (C,D)

        // A fragments (ISA 7.12.2, 32-bit A 16x4): VGPR0 = K=k0 (lo lanes) /
        // K=k0+2 (hi lanes), VGPR1 = K=k0+1 / K=k0+3. Shared by both N-tiles.
        v2f afrag[8];
        #pragma unroll
        for (int kk = 0; kk < 8; ++kk) {
            int k0 = kk * 4 + hi * 2;
            afrag[kk].x = pooled[mrow][k0];
            afrag[kk].y = pooled[mrow][k0 + 1];
        }

        // Hoist the empty-molecule predicate per output row (branchless later).
        float keep[8];
        #pragma unroll
        for (int r = 0; r < 8; ++r)
            keep[r] = (lenL[r + hi * 8] > 0) ? 1.0f : 0.0f;

        const bool full_tile = (m0 + MOLS_PER_WG) <= n_mols; // block-uniform

        #pragma unroll
        for (int t = 0; t < 2; ++t) {
            int n0 = (wave * 2 + t) * 16;
            int n  = n0 + mrow;
            float bv = bias[n];
            v8f c;
            #pragma unroll
            for (int i = 0; i < 8; ++i) c[i] = bv;

            const float* wrow = W + (size_t)n * PER_COL_DIM; // proj_w[n][*]
            #pragma unroll
            for (int kk = 0; kk < 8; ++kk) {
                int k0 = kk * 4 + hi * 2;
                v2f b;
                b.x = wrow[k0];       // B[k][n] = proj_w[n][k]
                b.y = wrow[k0 + 1];
                // 8 args: (neg_a, A, neg_b, B, c_mod, C, reuse_a, reuse_b)
                c = __builtin_amdgcn_wmma_f32_16x16x4_f32(
                        false, afrag[kk], false, b, (short)0, c, false, false);
            }

            // C/D layout: VGPR r, lanes 0-15 -> M=r; lanes 16-31 -> M=r+8.
            // keep[] zeroes empty molecules (reference emits exact 0, not bias).
            if (full_tile) {
                #pragma unroll
                for (int r = 0; r < 8; ++r) {
                    int m = m0 + r + hi * 8;
                    out[(size_t)m * HIDDEN + n] = c[r] * keep[r];
                }
            } else {
                #pragma unroll
                for (int r = 0; r < 8; ++r) {
                    int m = m0 + r + hi * 8;
                    if (m < n_mols)
                        out[(size_t)m * HIDDEN + n] = c[r] * keep[r];
                }
            }
        }
    }
}

extern "C" void kernel_launch(void* const* d_in, const int* in_sizes, int n_in,
                              void* d_out, int out_size, void* d_ws, size_t ws_size,
                              hipStream_t stream) {
    const int*   a_features = (const int*)d_in[0];
    const int*   a_scopes   = (const int*)d_in[1];
    const float* emb        = (const float*)d_in[2];
    const float* W          = (const float*)d_in[3];
    const float* b          = (const float*)d_in[4];
    float*       out        = (float*)d_out;

    int n_mols = in_sizes[1] / 2;
    dim3 grid((n_mols + MOLS_PER_WG - 1) / MOLS_PER_WG);
    GraphNodeFeatureEncoder_kernel<<<grid, WG_THREADS, 0, stream>>>(
        a_features, a_scopes, emb, W, b, out, n_mols);
}